// MambaLayer_47579647705759
// MI455X (gfx1250) — compile-verified
//
#include <hip/hip_runtime.h>
#include <math.h>
#include <stdint.h>

typedef __attribute__((ext_vector_type(2))) float v2f;
typedef __attribute__((ext_vector_type(8))) float v8f;

constexpr int LSEQ   = 16384;          // 32*32*16
constexpr int BSZ    = 2;
constexpr int BL     = BSZ * LSEQ;     // 32768 token rows
constexpr int CDIM   = 256;
constexpr int DIN    = 512;            // D_INNER
constexpr int XZW    = 1024;           // 2*D_INNER
constexpr int XDBLW  = 48;             // DT_RANK + 2*D_STATE
constexpr int DSTATE = 16;
constexpr int DTRANK = 16;
constexpr int NCHUNK = 256;            // scan chunks
constexpr int CLEN   = LSEQ / NCHUNK;  // 64 steps per chunk

__device__ __forceinline__ float silu_f(float v) { return v / (1.f + __expf(-v)); }

// gfx1250 async global->LDS copy (ASYNCcnt-tracked). VDST = LDS byte offset,
// VADDR = 64-bit global address.  (CDNA5 ISA)
__device__ __forceinline__ void async_copy_b32(uint32_t lds_off, const void* gptr) {
    asm volatile("global_load_async_to_lds_b32 %0, %1, off"
                 :: "v"(lds_off), "v"(gptr) : "memory");
}
__device__ __forceinline__ void wait_async0() {
    asm volatile("s_wait_asynccnt 0x0" ::: "memory");
}
__device__ __forceinline__ uint32_t lds_off_of(const void* shared_ptr) {
    // low 32 bits of a flat shared-aperture address == LDS byte offset
    return (uint32_t)(uintptr_t)shared_ptr;
}

// ---------------------------------------------------------------------------
// LayerNorm over C=256 for each (b,t) row of x laid out (B, C, L).
// One wave (32 lanes) per row, 8 elements per lane, shuffle reduction.
// ---------------------------------------------------------------------------
__global__ __launch_bounds__(256) void ln_kernel(const float* __restrict__ x,
                                                 const float* __restrict__ w,
                                                 const float* __restrict__ b,
                                                 float* __restrict__ xn) {
    const int row  = blockIdx.x * 8 + (threadIdx.x >> 5);   // [0, BL)
    const int lane = threadIdx.x & 31;
    const int bb   = row >> 14;                             // row / LSEQ
    const int t    = row & (LSEQ - 1);
    const float* xp = x + (size_t)bb * CDIM * LSEQ + t;     // element c at xp[c*LSEQ]

    float vals[8];
    float sum = 0.f;
#pragma unroll
    for (int i = 0; i < 8; ++i) {
        float v = xp[(size_t)(lane * 8 + i) * LSEQ];
        vals[i] = v;
        sum += v;
    }
#pragma unroll
    for (int off = 16; off > 0; off >>= 1) sum += __shfl_xor(sum, off, 32);
    const float mu = sum * (1.f / 256.f);

    float sq = 0.f;
#pragma unroll
    for (int i = 0; i < 8; ++i) { float d = vals[i] - mu; sq += d * d; }
#pragma unroll
    for (int off = 16; off > 0; off >>= 1) sq += __shfl_xor(sq, off, 32);
    const float rstd = rsqrtf(sq * (1.f / 256.f) + 1e-5f);

#pragma unroll
    for (int i = 0; i < 8; ++i) {
        const int c = lane * 8 + i;
        xn[(size_t)row * CDIM + c] = (vals[i] - mu) * rstd * w[c] + b[c];
    }
}

// ---------------------------------------------------------------------------
// Generic f32 WMMA GEMM:  C(M x N) = A(M x K, row stride lda) * B(K x N)
// grid.x = M/16 (M-tile), grid.y covers N in 128-col blocks (8 waves * 16).
// Each wave computes one 16x16 tile with V_WMMA_F32_16X16X4_F32.
// A-tile (16 x K) staged in LDS, row stride K+4 (bank-conflict-free frags).
// ACT: 0 = none, 2 = softplus.  TSTORE: write result as (B, N, L) transposed.
// ---------------------------------------------------------------------------
template <int ACT, bool TSTORE>
__global__ __launch_bounds__(256) void gemm16(const float* __restrict__ A, int lda,
                                              const float* __restrict__ Bw,
                                              const float* __restrict__ bias,
                                              float* __restrict__ Cp, int ldc,
                                              int N, int K) {
    extern __shared__ float sA[];                 // 16 * (K+4) floats
    const int tid  = threadIdx.x;
    const int wave = tid >> 5;
    const int lane = tid & 31;
    const int m0   = blockIdx.x * 16;
    const int n0   = (blockIdx.y * 8 + wave) * 16;
    const int ldsk = K + 4;

    // Cooperative float4 load of the 16 x K A-tile into LDS.
    const int kvec = K >> 2;                      // float4s per row
    for (int i = tid; i < 16 * kvec; i += 256) {
        const int row = i / kvec;
        const int c4  = (i % kvec) << 2;
        const float4 v = *reinterpret_cast<const float4*>(A + (size_t)(m0 + row) * lda + c4);
        float* dst = &sA[row * ldsk + c4];
        dst[0] = v.x; dst[1] = v.y; dst[2] = v.z; dst[3] = v.w;
    }
    __syncthreads();
    if (n0 >= N) return;                          // tail waves (N not multiple of 128)

    // ISA f32 A/B fragment layout: lanes 0-15 -> K pair {0,1}, lanes 16-31 -> {2,3}
    const int mrow  = lane & 15;
    const int khalf = (lane >> 4) << 1;
    const int ncol  = n0 + (lane & 15);

    v8f acc = {};
    for (int k0 = 0; k0 < K; k0 += 4) {
        v2f a, b;
        a.x = sA[mrow * ldsk + k0 + khalf];
        a.y = sA[mrow * ldsk + k0 + khalf + 1];
        b.x = Bw[(size_t)(k0 + khalf) * N + ncol];
        b.y = Bw[(size_t)(k0 + khalf + 1) * N + ncol];
        acc = __builtin_amdgcn_wmma_f32_16x16x4_f32(false, a, false, b,
                                                    (short)0, acc, false, false);
    }

    const float bv = bias ? bias[ncol] : 0.f;
#pragma unroll
    for (int v = 0; v < 8; ++v) {
        const int m = m0 + v + ((lane >> 4) << 3);   // D layout: VGPR v -> M = v / v+8
        float r = acc[v] + bv;
        if (ACT == 2) r = (r > 20.f) ? r : log1pf(__expf(r));   // softplus
        if (TSTORE) {
            const int bb = m >> 14;                   // m / LSEQ
            const int l  = m & (LSEQ - 1);
            Cp[((size_t)bb * ldc + ncol) * LSEQ + l] = r;
        } else {
            Cp[(size_t)m * ldc + ncol] = r;
        }
    }
}

// ---------------------------------------------------------------------------
// Depthwise causal conv (taps=4) along L per channel, + bias, + SiLU.
// xs = xz[..., :512].  One thread per (b,t,d).
// ---------------------------------------------------------------------------
__global__ __launch_bounds__(256) void conv_silu(const float* __restrict__ xz,
                                                 const float* __restrict__ cw,
                                                 const float* __restrict__ cb,
                                                 float* __restrict__ xc) {
    const size_t idx = (size_t)blockIdx.x * 256 + threadIdx.x;   // over BL*DIN
    const int d  = (int)(idx & (DIN - 1));
    const size_t bt = idx >> 9;                                  // / DIN
    const int t  = (int)(bt & (LSEQ - 1));
    const size_t bbase = (bt - t);                               // b * LSEQ
    float acc = cb[d];
#pragma unroll
    for (int k = 0; k < 4; ++k) {
        const int tt = t - 3 + k;
        if (tt >= 0) acc += xz[(bbase + tt) * XZW + d] * cw[d * 4 + k];
    }
    xc[idx] = silu_f(acc);
}

// ---------------------------------------------------------------------------
// Chunk-parallel selective scan, pass 1:
// per (b, chunk, d): local recurrence over CLEN steps with h=0, emit final
// h[16] and sum(dt) (the chunk decay product is exp(A[d,s]*sum_dt)).
// Block = 512 threads (one per d) for one (b, chunk).  B_t staged via
// GLOBAL_LOAD_ASYNC_TO_LDS_B32.
// ---------------------------------------------------------------------------
__global__ __launch_bounds__(512) void scan_part1(const float* __restrict__ xdbl,
                                                  const float* __restrict__ dt,
                                                  const float* __restrict__ xc,
                                                  const float* __restrict__ A_log,
                                                  float* __restrict__ hfin,
                                                  float* __restrict__ dtsum) {
    const int chunk = blockIdx.x & (NCHUNK - 1);
    const int b     = blockIdx.x >> 8;
    const int d     = threadIdx.x;
    __shared__ float sB[CLEN][DSTATE];

    const size_t rowbase = (size_t)b * LSEQ + (size_t)chunk * CLEN;

    const uint32_t ldsB = lds_off_of(&sB[0][0]);
    for (int i = d; i < CLEN * DSTATE; i += 512) {
        const int tt = i >> 4, s = i & 15;
        async_copy_b32(ldsB + (uint32_t)i * 4u,
                       xdbl + (rowbase + tt) * XDBLW + DTRANK + s);
    }
    wait_async0();
    __syncthreads();

    float Ad[DSTATE];
#pragma unroll
    for (int s = 0; s < DSTATE; ++s) Ad[s] = -__expf(A_log[d * DSTATE + s]);

    float h[DSTATE];
#pragma unroll
    for (int s = 0; s < DSTATE; ++s) h[s] = 0.f;
    float dts = 0.f;

    for (int tt = 0; tt < CLEN; ++tt) {
        const size_t bt  = rowbase + tt;
        const float dtv  = dt[bt * DIN + d];
        const float dtxv = dtv * xc[bt * DIN + d];
        dts += dtv;
#pragma unroll
        for (int s = 0; s < DSTATE; ++s) {
            const float a = __expf(dtv * Ad[s]);
            h[s] = a * h[s] + dtxv * sB[tt][s];
        }
    }
    const size_t obase = ((size_t)b * NCHUNK + chunk) * DIN + d;
#pragma unroll
    for (int s = 0; s < DSTATE; ++s) hfin[obase * DSTATE + s] = h[s];
    dtsum[obase] = dts;
}

// ---------------------------------------------------------------------------
// Pass 2: serial combine across chunks.  One thread per (b, d, s) = 16384.
// Rewrites hfin[b,c,d,s] in place with the incoming state h_in for chunk c.
// ---------------------------------------------------------------------------
__global__ __launch_bounds__(256) void scan_combine(const float* __restrict__ A_log,
                                                    const float* __restrict__ dtsum,
                                                    float* __restrict__ hfin) {
    const int idx = blockIdx.x * 256 + threadIdx.x;      // [0, BSZ*DIN*DSTATE)
    const int s = idx & (DSTATE - 1);
    const int d = (idx >> 4) & (DIN - 1);
    const int b = idx >> 13;                             // / (DIN*DSTATE)
    const float Ad = -__expf(A_log[d * DSTATE + s]);
    float h = 0.f;
    for (int c = 0; c < NCHUNK; ++c) {
        const size_t base = ((size_t)b * NCHUNK + c) * DIN + d;
        const float fin = hfin[base * DSTATE + s];
        const float P   = __expf(Ad * dtsum[base]);      // chunk decay product
        hfin[base * DSTATE + s] = h;                     // becomes h_in for chunk c
        h = P * h + fin;
    }
}

// ---------------------------------------------------------------------------
// Pass 3: re-run each chunk seeded with h_in, emitting
// y = (scan + D*xc) * silu(z), written in-place over xc.
// ---------------------------------------------------------------------------
__global__ __launch_bounds__(512) void scan_part3(const float* __restrict__ xdbl,
                                                  const float* __restrict__ dt,
                                                  const float* __restrict__ xc,
                                                  const float* __restrict__ xz,
                                                  const float* __restrict__ A_log,
                                                  const float* __restrict__ Dp,
                                                  const float* __restrict__ hin,
                                                  float* __restrict__ y) {
    const int chunk = blockIdx.x & (NCHUNK - 1);
    const int b     = blockIdx.x >> 8;
    const int d     = threadIdx.x;
    __shared__ float sB[CLEN][DSTATE];
    __shared__ float sC[CLEN][DSTATE];

    const size_t rowbase = (size_t)b * LSEQ + (size_t)chunk * CLEN;

    const uint32_t ldsB = lds_off_of(&sB[0][0]);
    const uint32_t ldsC = lds_off_of(&sC[0][0]);
    for (int i = d; i < CLEN * DSTATE; i += 512) {
        const int tt = i >> 4, s = i & 15;
        const float* row = xdbl + (rowbase + tt) * XDBLW + DTRANK;
        async_copy_b32(ldsB + (uint32_t)i * 4u, row + s);
        async_copy_b32(ldsC + (uint32_t)i * 4u, row + DSTATE + s);
    }
    wait_async0();
    __syncthreads();

    float Ad[DSTATE];
#pragma unroll
    for (int s = 0; s < DSTATE; ++s) Ad[s] = -__expf(A_log[d * DSTATE + s]);
    const float Dd = Dp[d];

    const size_t ibase = (((size_t)b * NCHUNK + chunk) * DIN + d) * DSTATE;
    float h[DSTATE];
#pragma unroll
    for (int s = 0; s < DSTATE; ++s) h[s] = hin[ibase + s];

    for (int tt = 0; tt < CLEN; ++tt) {
        const size_t bt  = rowbase + tt;
        const float dtv  = dt[bt * DIN + d];
        const float xv   = xc[bt * DIN + d];
        const float dtxv = dtv * xv;
        float yv = 0.f;
#pragma unroll
        for (int s = 0; s < DSTATE; ++s) {
            const float a = __expf(dtv * Ad[s]);
            h[s] = a * h[s] + dtxv * sB[tt][s];
            yv  += h[s] * sC[tt][s];
        }
        const float zv = xz[bt * XZW + DIN + d];
        y[bt * DIN + d] = (yv + Dd * xv) * silu_f(zv);
    }
}

// ---------------------------------------------------------------------------
extern "C" void kernel_launch(void* const* d_in, const int* in_sizes, int n_in,
                              void* d_out, int out_size, void* d_ws, size_t ws_size,
                              hipStream_t stream) {
    const float* x         = (const float*)d_in[0];
    const float* norm_w    = (const float*)d_in[1];
    const float* norm_b    = (const float*)d_in[2];
    const float* in_proj_w = (const float*)d_in[3];   // (256, 1024)
    const float* conv_w    = (const float*)d_in[4];   // (512, 4)
    const float* conv_b    = (const float*)d_in[5];
    const float* x_proj_w  = (const float*)d_in[6];   // (512, 48)
    const float* dt_proj_w = (const float*)d_in[7];   // (16, 512)
    const float* dt_proj_b = (const float*)d_in[8];
    const float* A_log     = (const float*)d_in[9];   // (512, 16)
    const float* Dvec      = (const float*)d_in[10];
    const float* out_proj_w= (const float*)d_in[11];  // (512, 256)
    float* out = (float*)d_out;

    // Workspace layout (floats)
    float* ws    = (float*)d_ws;
    float* xn    = ws;                                  // BL * 256
    float* xz    = xn    + (size_t)BL * CDIM;           // BL * 1024
    float* xc    = xz    + (size_t)BL * XZW;            // BL * 512 (reused as y)
    float* xdbl  = xc    + (size_t)BL * DIN;            // BL * 48
    float* dt    = xdbl  + (size_t)BL * XDBLW;          // BL * 512
    float* hfin  = dt    + (size_t)BL * DIN;            // BSZ*NCHUNK*DIN*16
    float* dtsum = hfin  + (size_t)BSZ * NCHUNK * DIN * DSTATE;  // BSZ*NCHUNK*DIN

    // 1. LayerNorm (transpose + normalize)
    ln_kernel<<<BL / 8, 256, 0, stream>>>(x, norm_w, norm_b, xn);

    // 2. in_proj: (32768 x 256) * (256 x 1024) -> xz
    gemm16<0, false><<<dim3(BL / 16, XZW / 128), 256,
                       16 * (CDIM + 4) * sizeof(float), stream>>>(
        xn, CDIM, in_proj_w, nullptr, xz, XZW, XZW, CDIM);

    // 3. depthwise causal conv + SiLU -> xc
    conv_silu<<<(size_t)BL * DIN / 256, 256, 0, stream>>>(xz, conv_w, conv_b, xc);

    // 4. x_proj: (32768 x 512) * (512 x 48) -> x_dbl
    gemm16<0, false><<<dim3(BL / 16, 1), 256,
                       16 * (DIN + 4) * sizeof(float), stream>>>(
        xc, DIN, x_proj_w, nullptr, xdbl, XDBLW, XDBLW, DIN);

    // 5. dt_proj + softplus: (32768 x 16) * (16 x 512) + b -> dt
    gemm16<2, false><<<dim3(BL / 16, DIN / 128), 256,
                       16 * (DTRANK + 4) * sizeof(float), stream>>>(
        xdbl, XDBLW, dt_proj_w, dt_proj_b, dt, DIN, DIN, DTRANK);

    // 6. chunk-parallel selective scan (3 passes) + D*xc + SiLU(z) gate
    scan_part1<<<BSZ * NCHUNK, DIN, 0, stream>>>(xdbl, dt, xc, A_log, hfin, dtsum);
    scan_combine<<<(BSZ * DIN * DSTATE) / 256, 256, 0, stream>>>(A_log, dtsum, hfin);
    scan_part3<<<BSZ * NCHUNK, DIN, 0, stream>>>(xdbl, dt, xc, xz, A_log, Dvec,
                                                 hfin, xc);

    // 7. out_proj with transposed store: (32768 x 512) * (512 x 256) -> (B, 256, L)
    gemm16<0, true><<<dim3(BL / 16, CDIM / 128), 256,
                      16 * (DIN + 4) * sizeof(float), stream>>>(
        xc, DIN, out_proj_w, nullptr, out, CDIM, CDIM, DIN);
}